// BatchedNMS_24249385353904
// MI455X (gfx1250) — compile-verified
//
#include <hip/hip_runtime.h>
#include <stdint.h>

#define N 8192
#define WORDS (N / 32)
#define IOU_THR 0.45f

typedef float v2f __attribute__((ext_vector_type(2)));
typedef float v8f __attribute__((ext_vector_type(8)));
typedef unsigned long long u64;
typedef unsigned int u32;

// ---- workspace byte offsets (total ~8.8 MB; mask has 4 padding rows for the
//      scan kernel's 4-deep load pipeline) ----
#define WS_MAXC   0u
#define WS_BOXOFF 256u                      // float4[N]  offset boxes
#define WS_AREA   (WS_BOXOFF + N * 16u)     // float[N]   areas (orig order)
#define WS_ORDER  (WS_AREA + N * 4u)        // int[N]     score-desc order
#define WS_SBOX   (WS_ORDER + N * 4u)       // float4[N]  sorted offset boxes
#define WS_SAREA  (WS_SBOX + N * 16u)       // float[N]   sorted areas
#define WS_SCAFT  (WS_SAREA + N * 4u)       // float[N]   scores after NMS (orig order)
#define WS_SUP    (WS_SCAFT + N * 4u)       // u32[WORDS] final suppression bitmap
#define WS_MASK   (WS_SUP + WORDS * 4u)     // u32[(N+4)*WORDS] suppression bit matrix

// ---------------------------------------------------------------------------
// Kernel 1: global max over all 4*N coordinates (single block).
// ---------------------------------------------------------------------------
__global__ __launch_bounds__(1024)
void nms_max_kernel(const float* __restrict__ boxes, float* __restrict__ maxc) {
    __shared__ float red[1024];
    int t = threadIdx.x;
    float m = 0.0f;  // all coords are >= 0
    for (int i = t; i < N * 4; i += 1024) m = fmaxf(m, boxes[i]);
    red[t] = m;
    __syncthreads();
    for (int s = 512; s > 0; s >>= 1) {
        if (t < s) red[t] = fmaxf(red[t], red[t + s]);
        __syncthreads();
    }
    if (t == 0) *maxc = red[0];
}

// ---------------------------------------------------------------------------
// Kernel 2: per-class offset boxes + areas.
// ---------------------------------------------------------------------------
__global__ __launch_bounds__(256)
void nms_prepare_kernel(const float4* __restrict__ boxes,
                        const int* __restrict__ idxs,
                        const float* __restrict__ maxc,
                        float4* __restrict__ boxoff,
                        float* __restrict__ area) {
    int i = blockIdx.x * blockDim.x + threadIdx.x;
    if (i >= N) return;
    float off = (float)idxs[i] * (*maxc + 1.0f);
    float4 b = boxes[i];
    float4 bo;
    bo.x = b.x + off; bo.y = b.y + off; bo.z = b.z + off; bo.w = b.w + off;
    boxoff[i] = bo;
    area[i] = (b.z - b.x) * (b.w - b.y);
}

// ---------------------------------------------------------------------------
// In-LDS bitonic sort of N u64 keys, ascending. 1024 threads.
// ---------------------------------------------------------------------------
__device__ __forceinline__ void bitonic_sort_lds(u64* s, int tid) {
    for (int k = 2; k <= N; k <<= 1) {
        for (int j = k >> 1; j > 0; j >>= 1) {
            __syncthreads();
            for (int i = tid; i < N; i += 1024) {
                int ixj = i ^ j;
                if (ixj > i) {
                    bool up = ((i & k) == 0);
                    u64 a = s[i], b = s[ixj];
                    if ((a > b) == up) { s[i] = b; s[ixj] = a; }
                }
            }
        }
    }
    __syncthreads();
}

// ---------------------------------------------------------------------------
// Kernel 3: argsort by (score desc, idx asc); gather sorted boxes/areas.
// Key = (~score_bits << 32) | idx  (scores are positive floats -> monotonic).
// ---------------------------------------------------------------------------
__global__ __launch_bounds__(1024)
void nms_sort1_kernel(const float* __restrict__ scores,
                      const float4* __restrict__ boxoff,
                      const float* __restrict__ area,
                      int* __restrict__ order,
                      float4* __restrict__ sbox,
                      float* __restrict__ sarea) {
    __shared__ u64 key[N];
    int t = threadIdx.x;
    for (int i = t; i < N; i += 1024) {
        u32 sb = __float_as_uint(scores[i]);
        key[i] = ((u64)(0xFFFFFFFFu - sb) << 32) | (u32)i;
    }
    bitonic_sort_lds(key, t);
    for (int k = t; k < N; k += 1024) {
        int idx = (int)(key[k] & 0xFFFFFFFFu);
        order[k] = idx;
        sbox[k]  = boxoff[idx];
        sarea[k] = area[idx];
    }
}

// ---------------------------------------------------------------------------
// Kernel 4: suppression bit-matrix. One wave per 16-row x 1024-column tile (8 waves per stripe).
// area_i + area_j outer-sum via V_WMMA_F32_16X16X4_F32; per-row 16-bit masks
// assembled with wave32 ballots (D layout: row = v + 8*(lane>=16), col=lane&15).
// Corner min/max on raw bits (non-negative floats: IEEE order == u32 order).
// Division-free overlap test: inter/denom > thr <=> inter > thr*denom.
// ---------------------------------------------------------------------------
__device__ __forceinline__ u32 umx(u32 a, u32 b) { return a > b ? a : b; }
__device__ __forceinline__ u32 umn(u32 a, u32 b) { return a < b ? a : b; }

#define NMS_TILE(JB, BAL)                                                     \
    do {                                                                      \
        const int jb = (JB);                                                  \
        float aj = sarea[jb + c];                                             \
        v2f bop;                                                              \
        bop.x = lo ? 1.0f : 0.0f;                                             \
        bop.y = lo ? aj : 0.0f;                                               \
        v8f d = __builtin_amdgcn_wmma_f32_16x16x4_f32(                        \
            false, aop, false, bop, (short)0, cz, false, false);              \
        uint4 cb = sboxu[jb + c];                                             \
        const int jIdx = jb + c;                                              \
        _Pragma("unroll")                                                     \
        for (int v = 0; v < 8; ++v) {                                         \
            u32 xx1 = umx(rb[v].x, cb.x);                                     \
            u32 yy1 = umx(rb[v].y, cb.y);                                     \
            u32 xx2 = umn(rb[v].z, cb.z);                                     \
            u32 yy2 = umn(rb[v].w, cb.w);                                     \
            float iw = __uint_as_float(xx2) - __uint_as_float(xx1);           \
            float ih = __uint_as_float(yy2) - __uint_as_float(yy1);           \
            iw = fmaxf(iw, 0.0f);                                             \
            ih = fmaxf(ih, 0.0f);                                             \
            float inter = iw * ih;                                            \
            float denom = d[v] - inter;                                       \
            bool pred = (jIdx > (r0 + v + hi8)) && (inter > IOU_THR * denom); \
            BAL[v] = __builtin_amdgcn_ballot_w32(pred);                       \
        }                                                                     \
    } while (0)

__global__ __launch_bounds__(32)
void nms_mask_kernel(const float4* __restrict__ sbox,
                     const float* __restrict__ sarea,
                     u32* __restrict__ mask) {
    const uint4* __restrict__ sboxu = (const uint4*)sbox;
    const int lane = threadIdx.x;
    const int r0 = blockIdx.x * 16;
    const bool lo = lane < 16;
    const int c = lane & 15;
    const int hi8 = lo ? 0 : 8;

    uint4 rb[8];
#pragma unroll
    for (int v = 0; v < 8; ++v) rb[v] = sboxu[r0 + v + hi8];

    float ai = sarea[r0 + c];
    v2f aop;
    aop.x = lo ? ai : 0.0f;
    aop.y = lo ? 1.0f : 0.0f;
    v8f cz = {};

    const u32 sel = (u32)(lane & 7);
    const int wbeg = blockIdx.y * (WORDS / 8);
    const int wend = wbeg + (WORDS / 8);
    for (int wg = wbeg; wg < wend; wg += 4) {
        __builtin_prefetch(&sbox[wg * 32 + 128], 0, 0);
        u32 wv[4];
#pragma unroll
        for (int u = 0; u < 4; ++u) {
            const int w = wg + u;
            const int j0 = w * 32;
            u32 bal0[8], bal1[8];
            NMS_TILE(j0,      bal0);
            NMS_TILE(j0 + 16, bal1);
            u32 b0 = bal0[0], b1 = bal1[0];
#pragma unroll
            for (int q = 1; q < 8; ++q) {
                b0 = (sel == (u32)q) ? bal0[q] : b0;
                b1 = (sel == (u32)q) ? bal1[q] : b1;
            }
            u32 m0 = (lane < 8) ? (b0 & 0xFFFFu) : (b0 >> 16);
            u32 m1 = (lane < 8) ? (b1 & 0xFFFFu) : (b1 >> 16);
            wv[u] = m0 | (m1 << 16);
        }
        if (lane < 16) {
            uint4 pack;
            pack.x = wv[0]; pack.y = wv[1]; pack.z = wv[2]; pack.w = wv[3];
            *(uint4*)&mask[(size_t)(r0 + lane) * WORDS + wg] = pack;
        }
    }
}

// ---------------------------------------------------------------------------
// Kernel 5: greedy sequential scan on a SINGLE wave32 — zero barriers.
// Lane l owns suppression words 8l..8l+7 in 8 VGPRs. The decision bit for box
// i is one v_readlane (uniform lane/word -> pure SALU branch). Mask rows are
// consumed strictly in order, so they are unconditionally software-pipelined
// 4 rows deep (2x b128 per lane per row, coalesced 1KB rows). Requires 4
// padding rows after the mask matrix.
// ---------------------------------------------------------------------------
__global__ __launch_bounds__(32)
void nms_scan_kernel(const u32* __restrict__ mask, u32* __restrict__ supg) {
    const int lane = threadIdx.x;
    const u32* base = mask + lane * 8;

    u32 sup[8];
#pragma unroll
    for (int r = 0; r < 8; ++r) sup[r] = 0u;

    uint4 bufA[4], bufB[4];
#pragma unroll
    for (int p = 0; p < 4; ++p) {
        bufA[p] = *(const uint4*)(base + (size_t)p * WORDS);
        bufB[p] = *(const uint4*)(base + (size_t)p * WORDS + 4);
    }

    int i = 0;
    for (int lown = 0; lown < 32; ++lown) {       // owner lane of current word
#pragma unroll
        for (int reg = 0; reg < 8; ++reg) {       // owner's register (word = lown*8+reg)
            for (int b4 = 0; b4 < 32; b4 += 4) {  // bit within word, 4 at a time
#pragma unroll
                for (int p = 0; p < 4; ++p) {
                    const int b = b4 + p;
                    u32 cur = (u32)__builtin_amdgcn_readlane((int)sup[reg], lown);
                    if (((cur >> b) & 1u) == 0u) {  // box i kept -> OR its row
                        sup[0] |= bufA[p].x; sup[1] |= bufA[p].y;
                        sup[2] |= bufA[p].z; sup[3] |= bufA[p].w;
                        sup[4] |= bufB[p].x; sup[5] |= bufB[p].y;
                        sup[6] |= bufB[p].z; sup[7] |= bufB[p].w;
                    }
                    // refill slot p with row i+4 (padding rows keep this in-bounds)
                    bufA[p] = *(const uint4*)(base + (size_t)(i + 4) * WORDS);
                    bufB[p] = *(const uint4*)(base + (size_t)(i + 4) * WORDS + 4);
                    ++i;
                }
            }
        }
    }

#pragma unroll
    for (int r = 0; r < 8; ++r) supg[lane * 8 + r] = sup[r];
}

// ---------------------------------------------------------------------------
// Kernel 5b: wide scatter of post-NMS scores back to original order.
// ---------------------------------------------------------------------------
__global__ __launch_bounds__(256)
void nms_scatter_kernel(const u32* __restrict__ supg,
                        const int* __restrict__ order,
                        const float* __restrict__ scores,
                        float* __restrict__ scaft) {
    int k = blockIdx.x * blockDim.x + threadIdx.x;
    if (k >= N) return;
    bool suppd = (supg[k >> 5] >> (k & 31)) & 1u;
    int orig = order[k];
    scaft[orig] = suppd ? 0.0f : scores[orig];
}

// ---------------------------------------------------------------------------
// Kernel 6: final stable argsort of -scores_after; emit [boxes|score] rows
// and int32 indices.
// ---------------------------------------------------------------------------
__global__ __launch_bounds__(1024)
void nms_final_kernel(const float* __restrict__ scaft,
                      const float4* __restrict__ boxes,
                      float* __restrict__ out) {
    __shared__ u64 key[N];
    int t = threadIdx.x;
    for (int i = t; i < N; i += 1024) {
        u32 sb = __float_as_uint(scaft[i]);
        key[i] = ((u64)(0xFFFFFFFFu - sb) << 32) | (u32)i;
    }
    bitonic_sort_lds(key, t);
    int* inds = (int*)out + N * 5;
    for (int k = t; k < N; k += 1024) {
        int idx = (int)(key[k] & 0xFFFFFFFFu);
        float4 b = boxes[idx];
        out[k * 5 + 0] = b.x;
        out[k * 5 + 1] = b.y;
        out[k * 5 + 2] = b.z;
        out[k * 5 + 3] = b.w;
        out[k * 5 + 4] = scaft[idx];
        inds[k] = idx;
    }
}

// ---------------------------------------------------------------------------
extern "C" void kernel_launch(void* const* d_in, const int* in_sizes, int n_in,
                              void* d_out, int out_size, void* d_ws, size_t ws_size,
                              hipStream_t stream) {
    const float* boxes  = (const float*)d_in[0];
    const float* scores = (const float*)d_in[1];
    const int*   idxs   = (const int*)d_in[2];
    (void)in_sizes; (void)n_in; (void)out_size; (void)ws_size;

    char* ws = (char*)d_ws;
    float*  maxc   = (float*)(ws + WS_MAXC);
    float4* boxoff = (float4*)(ws + WS_BOXOFF);
    float*  area   = (float*)(ws + WS_AREA);
    int*    order  = (int*)(ws + WS_ORDER);
    float4* sbox   = (float4*)(ws + WS_SBOX);
    float*  sarea  = (float*)(ws + WS_SAREA);
    float*  scaft  = (float*)(ws + WS_SCAFT);
    u32*    supg   = (u32*)(ws + WS_SUP);
    u32*    mask   = (u32*)(ws + WS_MASK);

    nms_max_kernel<<<1, 1024, 0, stream>>>(boxes, maxc);
    nms_prepare_kernel<<<N / 256, 256, 0, stream>>>((const float4*)boxes, idxs, maxc,
                                                    boxoff, area);
    nms_sort1_kernel<<<1, 1024, 0, stream>>>(scores, boxoff, area, order, sbox, sarea);
    nms_mask_kernel<<<dim3(N / 16, 8), 32, 0, stream>>>(sbox, sarea, mask);
    nms_scan_kernel<<<1, 32, 0, stream>>>(mask, supg);
    nms_scatter_kernel<<<N / 256, 256, 0, stream>>>(supg, order, scores, scaft);
    nms_final_kernel<<<1, 1024, 0, stream>>>(scaft, (const float4*)boxes, (float*)d_out);
}